// swin_infonce_region_cluster_22789096473168
// MI455X (gfx1250) — compile-verified
//
#include <hip/hip_runtime.h>
#include <hip/hip_bf16.h>
#include <math.h>

// ---------------------------------------------------------------------------
// Problem constants (from reference):
//   B=8, dim=96 (4 heads x 24), H=W=128 -> HW=16384 pixels/batch
//   FOLD=2, PROP=2 -> 128 regions of 64x64 points, 4 centers each
// ---------------------------------------------------------------------------
#define HW      16384
#define NBATCH  8

typedef float v2f __attribute__((ext_vector_type(2)));
typedef float v8f __attribute__((ext_vector_type(8)));

// ---------------------------------------------------------------------------
// Fused feat+value 1x1 conv GEMM:  fv[oc,p] = sum_k Wcat[oc,k]*x[k,p] + bcat[oc]
// with Wcat = [f_w ; v_w]  (192 x 96), per batch, V_WMMA_F32_16X16X4_F32.
// Block = 256 threads (8 waves). Each wave owns one 16-pixel N tile and all
// 12 M tiles (192 out channels) -> each x tile is loaded once and reused for
// both feat and value (halves HBM reads of x vs. two separate launches).
// Weights staged in dynamic LDS (72 KB + bias). Grid = NBATCH*(HW/128)=1024.
// ---------------------------------------------------------------------------
__global__ __launch_bounds__(256) void k_featval(
    const float* __restrict__ x,     // [B,96,HW]
    const float* __restrict__ f_w,   // [96,96]
    const float* __restrict__ f_b,   // [96]
    const float* __restrict__ v_w,   // [96,96]
    const float* __restrict__ v_b,   // [96]
    float* __restrict__ fv)          // [B,192,HW]  (feat: 0-95, value: 96-191)
{
  extern __shared__ float smem[];          // 192*96 weights + 192 bias
  float* sW = smem;                        // [192][96]
  float* sB = smem + 192 * 96;             // [192]

  const int t = threadIdx.x;
  for (int i = t; i < 96 * 96; i += 256) { sW[i] = f_w[i]; sW[96 * 96 + i] = v_w[i]; }
  if (t < 96) { sB[t] = f_b[t]; sB[96 + t] = v_b[t]; }
  __syncthreads();

  const int batch  = blockIdx.x >> 7;                             // /128
  const int pix0   = ((blockIdx.x & 127) << 7) + ((t >> 5) << 4); // block pix + wave*16
  const int lane   = t & 31;
  const int laneLo = lane & 15;
  const int laneHi = lane >> 4;

  const float* inb = x + (size_t)batch * 96 * HW;

  v8f acc[12];
  const v8f zero = {0.f, 0.f, 0.f, 0.f, 0.f, 0.f, 0.f, 0.f};
#pragma unroll
  for (int m = 0; m < 12; ++m) acc[m] = zero;

  // K = 96 in 24 steps of 4.
  // A (16x4 f32): lanes 0-15 hold M rows; VGPR0 = K {0 | 2 by lane half},
  //               VGPR1 = K {1 | 3}.  B (4x16): same K split, N = laneLo.
  for (int ks = 0; ks < 24; ++ks) {
    const int k0 = ks * 4 + 2 * laneHi;
    v2f bm;
    bm.x = inb[(size_t)k0 * HW + pix0 + laneLo];
    bm.y = inb[(size_t)(k0 + 1) * HW + pix0 + laneLo];
#pragma unroll
    for (int mt = 0; mt < 12; ++mt) {
      v2f am;
      am.x = sW[(mt * 16 + laneLo) * 96 + k0];
      am.y = sW[(mt * 16 + laneLo) * 96 + k0 + 1];
      acc[mt] = __builtin_amdgcn_wmma_f32_16x16x4_f32(
          false, am, false, bm, (short)0, acc[mt], false, false);
    }
  }

  // D layout: VGPR r -> M = r + 8*laneHi, N = laneLo
  float* outb = fv + (size_t)batch * 192 * HW;
#pragma unroll
  for (int mt = 0; mt < 12; ++mt) {
#pragma unroll
    for (int r = 0; r < 8; ++r) {
      const int oc = mt * 16 + r + 8 * laneHi;
      outb[(size_t)oc * HW + pix0 + laneLo] = acc[mt][r] + sB[oc];
    }
  }
}

// ---------------------------------------------------------------------------
// 96->96 1x1 conv GEMM (used for proj): out[oc,p] = sum_k W[oc,k]*in[k,p]+b[oc]
// Same structure, 6 M tiles, static 36 KB LDS.
// ---------------------------------------------------------------------------
__global__ __launch_bounds__(256) void k_conv96(
    const float* __restrict__ in,    // [B,96,HW]
    const float* __restrict__ W,     // [96,96] row-major [oc][ic]
    const float* __restrict__ bias,  // [96]
    float* __restrict__ out)         // [B,96,HW]
{
  __shared__ float sW[96 * 96];
  __shared__ float sB[96];
  const int t = threadIdx.x;
  for (int i = t; i < 96 * 96; i += 256) sW[i] = W[i];
  if (t < 96) sB[t] = bias[t];
  __syncthreads();

  const int batch  = blockIdx.x >> 7;
  const int pix0   = ((blockIdx.x & 127) << 7) + ((t >> 5) << 4);
  const int lane   = t & 31;
  const int laneLo = lane & 15;
  const int laneHi = lane >> 4;

  const float* inb = in + (size_t)batch * 96 * HW;

  v8f acc[6];
  const v8f zero = {0.f, 0.f, 0.f, 0.f, 0.f, 0.f, 0.f, 0.f};
#pragma unroll
  for (int m = 0; m < 6; ++m) acc[m] = zero;

  for (int ks = 0; ks < 24; ++ks) {
    const int k0 = ks * 4 + 2 * laneHi;
    v2f bm;
    bm.x = inb[(size_t)k0 * HW + pix0 + laneLo];
    bm.y = inb[(size_t)(k0 + 1) * HW + pix0 + laneLo];
#pragma unroll
    for (int mt = 0; mt < 6; ++mt) {
      v2f am;
      am.x = sW[(mt * 16 + laneLo) * 96 + k0];
      am.y = sW[(mt * 16 + laneLo) * 96 + k0 + 1];
      acc[mt] = __builtin_amdgcn_wmma_f32_16x16x4_f32(
          false, am, false, bm, (short)0, acc[mt], false, false);
    }
  }

  float* outb = out + (size_t)batch * 96 * HW;
#pragma unroll
  for (int mt = 0; mt < 6; ++mt) {
#pragma unroll
    for (int r = 0; r < 8; ++r) {
      const int oc = mt * 16 + r + 8 * laneHi;
      outb[(size_t)oc * HW + pix0 + laneLo] = acc[mt][r] + sB[oc];
    }
  }
}

// ---------------------------------------------------------------------------
// Region clustering. One 256-thread workgroup per region (128 regions).
// Region br: f2=br&1, f1=(br>>1)&1, head=(br>>2)&3, batch=br>>4.
// Each thread owns 16 points of one quadrant (64 threads per quadrant).
// fv buffer layout: [B,192,HW], feat = ch 0-95, value = ch 96-191.
// ---------------------------------------------------------------------------
__global__ __launch_bounds__(256) void k_cluster(
    const float* __restrict__ fv,      // [B,192,HW]
    float* __restrict__ merged,        // [B,96,HW]
    const float* __restrict__ alpha_p, // [1]
    const float* __restrict__ beta_p)  // [1]
{
  __shared__ float s_csum[4][24];   // feat pooled sums  (per center)
  __shared__ float s_vsum[4][24];   // value pooled sums
  __shared__ float s_ctrn[4][24];   // l2-normalized centers
  __shared__ float s_vc[4][24];     // value-center means
  __shared__ float s_aggn[4][24];   // agg numerator
  __shared__ float s_agg[4][24];    // final agg
  __shared__ float s_den[4];        // agg denominator (sum of masked sim)

  const int br    = blockIdx.x;
  const int f2    = br & 1;
  const int f1    = (br >> 1) & 1;
  const int head  = (br >> 2) & 3;
  const int batch = br >> 4;

  const int t   = threadIdx.x;
  const int m_q = t >> 6;              // quadrant this thread pools
  const int q   = t & 63;
  const int qi  = (m_q >> 1) * 32;     // quadrant origin in region coords
  const int qj  = (m_q & 1) * 32;

  const float* featp = fv + ((size_t)batch * 192 + head * 24) * HW;
  const float* valp  = featp + (size_t)96 * HW;
  float* outp = merged + ((size_t)batch * 96 + head * 24) * HW;
  const int rbase = f1 * 64, cbase = f2 * 64;

  if (t < 96) {
    s_csum[t / 24][t % 24] = 0.f;
    s_vsum[t / 24][t % 24] = 0.f;
    s_aggn[t / 24][t % 24] = 0.f;
  }
  if (t < 4) s_den[t] = 0.f;
  __syncthreads();

  // ---- pass 1: adaptive-avg-pool (32x32 block sums per quadrant) ----
  float acc_c[24], acc_v[24];
#pragma unroll
  for (int ch = 0; ch < 24; ++ch) { acc_c[ch] = 0.f; acc_v[ch] = 0.f; }
  for (int p = 0; p < 16; ++p) {
    const int flat = p * 64 + q;                  // 0..1023 within quadrant
    const int i = qi + (flat >> 5);
    const int j = qj + (flat & 31);
    const int pix = (rbase + i) * 128 + (cbase + j);
#pragma unroll
    for (int ch = 0; ch < 24; ++ch) {
      acc_c[ch] += featp[(size_t)ch * HW + pix];
      acc_v[ch] += valp[(size_t)ch * HW + pix];
    }
  }
#pragma unroll
  for (int ch = 0; ch < 24; ++ch) {
    atomicAdd(&s_csum[m_q][ch], acc_c[ch]);
    atomicAdd(&s_vsum[m_q][ch], acc_v[ch]);
  }
  __syncthreads();

  // ---- centers: means + l2norm (one thread per center) ----
  if (t < 4) {
    float nrm = 0.f;
    for (int ch = 0; ch < 24; ++ch) {
      const float cm = s_csum[t][ch] * (1.f / 1024.f);
      s_ctrn[t][ch] = cm;
      s_vc[t][ch]   = s_vsum[t][ch] * (1.f / 1024.f);
      nrm += cm * cm;
    }
    const float inv = 1.f / fmaxf(sqrtf(nrm), 1e-12f);
    for (int ch = 0; ch < 24; ++ch) s_ctrn[t][ch] *= inv;
  }
  __syncthreads();

  const float alpha = alpha_p[0];
  const float beta  = beta_p[0];

  // ---- pass 2: sim -> hard argmax -> aggregate ----
  float ls[16];
  int   li[16];
  for (int p = 0; p < 16; ++p) {
    const int flat = p * 64 + q;
    const int i = qi + (flat >> 5);
    const int j = qj + (flat & 31);
    const int pix = (rbase + i) * 128 + (cbase + j);

    float f[24];
    float nrm2 = 0.f;
#pragma unroll
    for (int ch = 0; ch < 24; ++ch) {
      f[ch] = featp[(size_t)ch * HW + pix];
      nrm2 += f[ch] * f[ch];
    }
    const float invn = 1.f / fmaxf(sqrtf(nrm2), 1e-12f);

    float best = -1.f;
    int idx = 0;
#pragma unroll
    for (int m = 0; m < 4; ++m) {
      float d = 0.f;
#pragma unroll
      for (int ch = 0; ch < 24; ++ch) d += s_ctrn[m][ch] * f[ch];
      const float sim = 1.f / (1.f + __expf(-(beta + alpha * (d * invn))));
      if (sim > best) { best = sim; idx = m; }   // first-max, matches argmax
    }
    ls[p] = best;
    li[p] = idx;

    atomicAdd(&s_den[idx], best);
#pragma unroll
    for (int ch = 0; ch < 24; ++ch)
      atomicAdd(&s_aggn[idx][ch], best * valp[(size_t)ch * HW + pix]);
  }
  __syncthreads();

  if (t < 96) {
    const int m = t / 24, ch = t % 24;
    s_agg[m][ch] = (s_aggn[m][ch] + s_vc[m][ch]) / (s_den[m] + 1.f);
  }
  __syncthreads();

  // ---- pass 3: dispatch centers back to points ----
  for (int p = 0; p < 16; ++p) {
    const int flat = p * 64 + q;
    const int i = qi + (flat >> 5);
    const int j = qj + (flat & 31);
    const int pix = (rbase + i) * 128 + (cbase + j);
    const float s = ls[p];
    const int idx = li[p];
#pragma unroll
    for (int ch = 0; ch < 24; ++ch)
      outp[(size_t)ch * HW + pix] = s_agg[idx][ch] * s;
  }
}

// ---------------------------------------------------------------------------
// kernel_launch
//   d_in: x, f_w, f_b, v_w, v_b, proj_w, proj_b, sim_alpha, sim_beta
//   workspace: fv [B,192,HW] (100.7 MB) | merged [B,96,HW] (50.3 MB)
// ---------------------------------------------------------------------------
extern "C" void kernel_launch(void* const* d_in, const int* in_sizes, int n_in,
                              void* d_out, int out_size, void* d_ws, size_t ws_size,
                              hipStream_t stream) {
  const float* x      = (const float*)d_in[0];
  const float* f_w    = (const float*)d_in[1];
  const float* f_b    = (const float*)d_in[2];
  const float* v_w    = (const float*)d_in[3];
  const float* v_b    = (const float*)d_in[4];
  const float* proj_w = (const float*)d_in[5];
  const float* proj_b = (const float*)d_in[6];
  const float* alpha  = (const float*)d_in[7];
  const float* beta   = (const float*)d_in[8];
  float* out = (float*)d_out;

  float* fv     = (float*)d_ws;                                   // [8,192,16384]
  float* merged = (float*)((char*)d_ws + (size_t)NBATCH * 192 * HW * sizeof(float));

  const dim3 blk(256);
  const dim3 grid_conv(NBATCH * (HW / 128));   // 1024 blocks, 128 pixels each
  const dim3 grid_clu(128);                    // one block per region

  const size_t fv_lds = (size_t)(192 * 96 + 192) * sizeof(float); // 74,496 B

  k_featval<<<grid_conv, blk, fv_lds, stream>>>(x, f_w, f_b, v_w, v_b, fv);
  k_cluster<<<grid_clu, blk, 0, stream>>>(fv, merged, alpha, beta);
  k_conv96<<<grid_conv, blk, 0, stream>>>(merged, proj_w, proj_b, out);
}